// InstanceGCN_2843268349979
// MI455X (gfx1250) — compile-verified
//
#include <hip/hip_runtime.h>

constexpr int KH = 10;   // hidden feature size H

typedef __attribute__((ext_vector_type(2))) float v2f;
typedef __attribute__((ext_vector_type(8))) float v8f;

// ---------------------------------------------------------------------------
// fast f32 atomic add (lowers to global_atomic_add_f32 on gfx1250)
// ---------------------------------------------------------------------------
__device__ __forceinline__ void fatomic_add(float* p, float v) {
  unsafeAtomicAdd(p, v);
}

// ---------------------------------------------------------------------------
// One 16x16 output tile:  C(16x16) = A(16xK) * B(Kx16), f32 accumulate,
// via V_WMMA_F32_16X16X4_F32 in K-chunks of 4.  Fully branchless:
//  - rows >= nrowsC are clamped to a duplicate in-bounds row (their outputs
//    are garbage but callers never consume them)
//  - tail-K / padded-column elements are loaded from a clamped in-bounds
//    address and zeroed with a select (v_cndmask), never with exec masking
// VGPR layouts per CDNA5 ISA 7.12.2:
//   A 16x4 f32 : lanes 0-15 M=lane (K=+0,+1), lanes 16-31 M=lane-16 (K=+2,+3)
//   B 4x16 f32 : lanes hold N=lane%16; half selects K rows +0/+1 vs +2/+3
//   C/D        : VGPR r -> M = r + (lane<16?0:8), N = lane%16
// ---------------------------------------------------------------------------
template<int K, int NCOLS>
__device__ __forceinline__ v8f wmma_tile_t(const float* __restrict__ A, int lda,
                                           int nrowsC /*>=1*/,
                                           const float* __restrict__ B, int ldb)
{
  const int lane = threadIdx.x & 31;
  const int half = lane >> 4;
  const int m    = lane & 15;
  const int mr   = (m < nrowsC) ? m : (nrowsC - 1);   // clamped A row
  const int mn   = (m < NCOLS)  ? m : (NCOLS - 1);    // clamped B col
  const bool cv  = (m < NCOLS);
  const float* Ar = A + (size_t)mr * lda;
  v8f c = {0.f, 0.f, 0.f, 0.f, 0.f, 0.f, 0.f, 0.f};
#pragma unroll
  for (int k0 = 0; k0 < K; k0 += 4) {
    const int ka = k0 + half * 2;     // first K index this lane-half supplies
    v2f a, b;
    if (k0 + 3 < K) {                 // compile-time: whole chunk in-bounds
      a.x = Ar[ka];
      a.y = Ar[ka + 1];
      const float b0 = B[(size_t)ka * ldb + mn];
      const float b1 = B[(size_t)(ka + 1) * ldb + mn];
      b.x = cv ? b0 : 0.f;
      b.y = cv ? b1 : 0.f;
    } else {                          // compile-time: tail chunk, mask values
      const bool v0 = (ka     < K);
      const bool v1 = (ka + 1 < K);
      const int ca0 = v0 ? ka     : 0;
      const int ca1 = v1 ? ka + 1 : 0;
      const float a0 = Ar[ca0], a1 = Ar[ca1];
      const float b0 = B[(size_t)ca0 * ldb + mn];
      const float b1 = B[(size_t)ca1 * ldb + mn];
      a.x = v0 ? a0 : 0.f;
      a.y = v1 ? a1 : 0.f;
      b.x = (v0 && cv) ? b0 : 0.f;
      b.y = (v1 && cv) ? b1 : 0.f;
    }
#if defined(__HIP_DEVICE_COMPILE__)
    c = __builtin_amdgcn_wmma_f32_16x16x4_f32(false, a, false, b,
                                              (short)0, c, false, false);
#else
    (void)a; (void)b;                 // host semantic pass only
#endif
  }
  return c;
}

// ---------------------------------------------------------------------------
// zero-fill
// ---------------------------------------------------------------------------
__global__ void k_zero(float* __restrict__ p, size_t n) {
  size_t i      = (size_t)blockIdx.x * blockDim.x + threadIdx.x;
  size_t stride = (size_t)gridDim.x * blockDim.x;
  for (; i < n; i += stride) p[i] = 0.f;
}

// ---------------------------------------------------------------------------
// per-relation degree counting (float counters, exact < 2^24)
// edge arrays streamed once -> non-temporal loads
// ---------------------------------------------------------------------------
__global__ void k_deg(const int* __restrict__ src, const int* __restrict__ dst,
                      float* __restrict__ dout, float* __restrict__ din, int E) {
  int i      = blockIdx.x * blockDim.x + threadIdx.x;
  int stride = gridDim.x * blockDim.x;
  for (; i < E; i += stride) {
    const int s = __builtin_nontemporal_load(src + i);
    const int d = __builtin_nontemporal_load(dst + i);
    fatomic_add(&dout[s], 1.f);
    fatomic_add(&din[d], 1.f);
  }
}

// ---------------------------------------------------------------------------
// dense node transform via WMMA:
//   out[row] = relu( A[row,0:F] @ W[F,10] + biasScale*bias ) * rsqrt(max(degout[row],1))
// one wave per 16-row tile, 4 waves per block; branchless epilogue (masked
// stores only: bias hoisted with clamped index, degout index clamped)
// ---------------------------------------------------------------------------
template<int F>
__global__ void k_dense(const float* __restrict__ A, int n,
                        const float* __restrict__ W, const float* __restrict__ bias,
                        float biasScale, const float* __restrict__ degout,
                        float* __restrict__ out)
{
  const int wave = threadIdx.x >> 5;
  const int tile = blockIdx.x * (blockDim.x >> 5) + wave;
  const int row0 = tile * 16;
  if (row0 >= n) return;                       // wave-uniform exit
  int nrows = n - row0; if (nrows > 16) nrows = 16;

  v8f c = wmma_tile_t<F, KH>(A + (size_t)row0 * F, F, nrows, W, KH);

  const int lane = threadIdx.x & 31;
  const int half = lane >> 4, m = lane & 15;
  const bool cm  = (m < KH);
  const float bm = bias[cm ? m : KH - 1] * biasScale;   // hoisted, in-bounds
#pragma unroll
  for (int r = 0; r < 8; ++r) {
    const int row = half * 8 + r;
    const int gr  = row0 + (row < nrows ? row : nrows - 1);  // clamped row
    const float dg = degout[gr];
    const float v  = fmaxf(c[r] + bm, 0.f) * rsqrtf(fmaxf(dg, 1.f));
    if (cm && row < nrows)
      out[(size_t)(row0 + row) * KH + m] = v;
  }
}

// ---------------------------------------------------------------------------
// edge scatter with destination-degree norm folded per edge:
//   agg[dst] += x[src] * w * rsqrt(max(deg_in[dst],1))
// edge arrays non-temporal; feature row as 5 aligned float2 loads
// ---------------------------------------------------------------------------
__global__ void k_scatter(const int* __restrict__ src, const int* __restrict__ dst,
                          const float* __restrict__ w, const float* __restrict__ degin,
                          const float* __restrict__ x, float* __restrict__ agg, int E)
{
  int i      = blockIdx.x * blockDim.x + threadIdx.x;
  int stride = gridDim.x * blockDim.x;
  for (; i < E; i += stride) {
    const int   s  = __builtin_nontemporal_load(src + i);
    const int   d  = __builtin_nontemporal_load(dst + i);
    const float we = __builtin_nontemporal_load(w + i);
    const float sc = we * rsqrtf(fmaxf(degin[d], 1.f));
    const float2* xr = (const float2*)(x + (size_t)s * KH);   // rows 8B-aligned
    float* ar = agg + (size_t)d * KH;
#pragma unroll
    for (int f = 0; f < KH / 2; ++f) {
      const float2 p = xr[f];
      fatomic_add(&ar[2 * f + 0], p.x * sc);
      fatomic_add(&ar[2 * f + 1], p.y * sc);
    }
  }
}

// ---------------------------------------------------------------------------
// var finalize + 3-layer MLP head + mean-reduce, all chained through WMMA.
// D-layout -> A-layout transpose between matmuls goes through LDS.
// ---------------------------------------------------------------------------
__global__ void k_head(const float* __restrict__ aggv, int n,
                       const float* __restrict__ W2, const float* __restrict__ b2,
                       const float* __restrict__ O1, const float* __restrict__ ob1,
                       const float* __restrict__ O2, const float* __restrict__ ob2,
                       const float* __restrict__ O3, const float* __restrict__ ob3,
                       float* __restrict__ accum)
{
  __shared__ float lds[4][16][17];
  __shared__ float lsum[4][16];
  const int wave = threadIdx.x >> 5;
  const int tile = blockIdx.x * (blockDim.x >> 5) + wave;
  const int lane = threadIdx.x & 31;
  const int half = lane >> 4, m = lane & 15;
  const bool cm  = (m < KH);
  const int mb   = cm ? m : KH - 1;                     // clamped bias index
  const int row0 = tile * 16;
  int nrows = n - row0;
  nrows = nrows < 0 ? 0 : (nrows > 16 ? 16 : nrows);
  const int nrC = (nrows > 0) ? nrows : 1;              // keep reads in-bounds
  const float* Abase = aggv + (size_t)((nrows > 0) ? row0 : 0) * KH;
  float (*T)[17] = lds[wave];
  float* S = lsum[wave];

  auto writeT = [&](const v8f& cc, const float* bias, float bs) {
    const float bm = bias[mb] * bs;                     // hoisted, unconditional
#pragma unroll
    for (int r = 0; r < 8; ++r) {
      const int row = half * 8 + r;
      const float v = fmaxf(cc[r] + bm, 0.f);
      T[row][m] = cm ? v : 0.f;                         // unconditional ds_store
    }
  };

  // h_var = relu(agg_var @ W2 + b2)     (agg_var already deg_in-normalized)
  v8f c = wmma_tile_t<KH, KH>(Abase, KH, nrC, W2, KH);
  writeT(c, b2, 1.f);
  __syncthreads();
  // t1 = relu(h_var @ O1 + ob1)
  c = wmma_tile_t<KH, KH>(&T[0][0], 17, 16, O1, KH);
  writeT(c, ob1, 1.f);
  __syncthreads();
  // t2 = relu(t1 @ O2 + ob2)
  c = wmma_tile_t<KH, KH>(&T[0][0], 17, 16, O2, KH);
  writeT(c, ob2, 1.f);
  __syncthreads();
  // logit = t2 @ O3 + ob3   (no relu), single output column
  c = wmma_tile_t<KH, 1>(&T[0][0], 17, 16, O3, 1);
  {
    const float o3b = ob3[0];
    if (m == 0) {                                       // lanes 0 and 16 only
#pragma unroll
      for (int r = 0; r < 8; ++r) S[half * 8 + r] = c[r] + o3b;
    }
  }
  __syncthreads();
  if (lane == 0 && nrows > 0) {
    float s = 0.f;
    for (int i = 0; i < nrows; ++i) s += S[i];
    fatomic_add(accum, s);
  }
}

__global__ void k_final(const float* __restrict__ acc, float* __restrict__ out, float inv) {
  if (blockIdx.x == 0 && threadIdx.x == 0) out[0] = acc[0] * inv;
}

// ---------------------------------------------------------------------------
extern "C" void kernel_launch(void* const* d_in, const int* in_sizes, int n_in,
                              void* d_out, int out_size, void* d_ws, size_t ws_size,
                              hipStream_t stream) {
  const float* var_feats = (const float*)d_in[0];
  const float* soc_feats = (const float*)d_in[1];
  /* d_in[2] con_feats: unused in the live forward path */
  const int*   v2c_src = (const int*)d_in[3];
  const int*   v2c_dst = (const int*)d_in[4];
  const int*   c2v_src = (const int*)d_in[5];
  const int*   c2v_dst = (const int*)d_in[6];
  const int*   s2c_src = (const int*)d_in[7];
  const int*   s2c_dst = (const int*)d_in[8];
  const float* w_v2c   = (const float*)d_in[9];
  const float* w_c2v   = (const float*)d_in[10];
  const float* w_s2c   = (const float*)d_in[11];
  const float* var_W   = (const float*)d_in[12];
  const float* var_b   = (const float*)d_in[13];
  const float* soc_W   = (const float*)d_in[14];
  const float* soc_b   = (const float*)d_in[15];
  /* 16/17 con_W/con_b unused; 18/19 W1/b1 dead (results overwritten) */
  const float* W2  = (const float*)d_in[20];
  const float* b2  = (const float*)d_in[21];
  const float* O1  = (const float*)d_in[22];
  const float* ob1 = (const float*)d_in[23];
  const float* O2  = (const float*)d_in[24];
  const float* ob2 = (const float*)d_in[25];
  const float* O3  = (const float*)d_in[26];
  const float* ob3 = (const float*)d_in[27];

  const int NV = in_sizes[0] / 6;
  const int NS = in_sizes[1] / 6;
  const int NC = in_sizes[2] / 4;
  const int E  = in_sizes[3];
  const int ES = in_sizes[7];

  // ---- workspace layout (floats) ----
  float* ws = (float*)d_ws;
  size_t off = 0;
  float* deg_out_var = ws + off; off += NV;           // v2c src degrees
  float* deg_in_conV = ws + off; off += NC;           // v2c dst degrees
  float* deg_out_soc = ws + off; off += NS;           // s2c src degrees
  float* deg_in_conS = ws + off; off += NC;           // s2c dst degrees
  float* deg_out_con = ws + off; off += NC;           // c2v src degrees
  float* deg_in_var  = ws + off; off += NV;           // c2v dst degrees
  float* agg_con     = ws + off; off += (size_t)NC * KH;
  float* agg_var     = ws + off; off += (size_t)NV * KH;
  float* accum       = ws + off; off += 1;
  const size_t zcount = off;                          // region needing zeroing
  float* Xs_var      = ws + off; off += (size_t)NV * KH;
  float* Xs_soc      = ws + off; off += (size_t)NS * KH;
  float* hcon        = ws + off; off += (size_t)NC * KH;

  // ---- zero counters / accumulators (every call: deterministic) ----
  {
    int blk = (int)((zcount + 255) / 256);
    if (blk > 16384) blk = 16384;
    k_zero<<<blk, 256, 0, stream>>>(ws, zcount);
  }

  // ---- degrees ----
  k_deg<<<4096, 256, 0, stream>>>(v2c_src, v2c_dst, deg_out_var, deg_in_conV, E);
  k_deg<<<1024, 256, 0, stream>>>(s2c_src, s2c_dst, deg_out_soc, deg_in_conS, ES);
  k_deg<<<4096, 256, 0, stream>>>(c2v_src, c2v_dst, deg_out_con, deg_in_var, E);

  // ---- input MLPs, pre-scaled by out-degree norm ----
  {
    int blocks = (((NV + 15) / 16) + 3) / 4;
    k_dense<6><<<blocks, 128, 0, stream>>>(var_feats, NV, var_W, var_b, 1.f,
                                           deg_out_var, Xs_var);
    blocks = (((NS + 15) / 16) + 3) / 4;
    k_dense<6><<<blocks, 128, 0, stream>>>(soc_feats, NS, soc_W, soc_b, 1.f,
                                           deg_out_soc, Xs_soc);
  }

  // ---- scatter v2c + s2c into one con aggregation buffer ----
  k_scatter<<<8192, 256, 0, stream>>>(v2c_src, v2c_dst, w_v2c, deg_in_conV,
                                      Xs_var, agg_con, E);
  k_scatter<<<2048, 256, 0, stream>>>(s2c_src, s2c_dst, w_s2c, deg_in_conS,
                                      Xs_soc, agg_con, ES);

  // ---- con transform: relu(agg @ W2 + 2*b2), scaled by con out-degree norm ----
  {
    int blocks = (((NC + 15) / 16) + 3) / 4;
    k_dense<KH><<<blocks, 128, 0, stream>>>(agg_con, NC, W2, b2, 2.f,
                                            deg_out_con, hcon);
  }

  // ---- scatter c2v into var aggregation buffer ----
  k_scatter<<<8192, 256, 0, stream>>>(c2v_src, c2v_dst, w_c2v, deg_in_var,
                                      hcon, agg_var, E);

  // ---- var finalize + MLP head + mean reduction ----
  {
    int blocks = (((NV + 15) / 16) + 3) / 4;
    k_head<<<blocks, 128, 0, stream>>>(agg_var, NV, W2, b2,
                                       O1, ob1, O2, ob2, O3, ob3, accum);
  }
  k_final<<<1, 32, 0, stream>>>(accum, (float*)d_out, 1.f / (float)NV);
}